// Communication_55551107007213
// MI455X (gfx1250) — compile-verified
//
#include <hip/hip_runtime.h>

// Problem constants (from reference setup_inputs)
#define BB   8
#define NAG  8
#define CCH  2
#define HH   384
#define WW   384
#define HWP  (HH*WW)            // 147456
#define BNHW (BB*NAG*HWP)       // 9437184
#define LSTR 134                // LDS row stride (132 cols padded; keeps 8B align)

typedef __attribute__((ext_vector_type(2))) float v2f;
typedef __attribute__((ext_vector_type(8))) float v8f;

// ---------------------------------------------------------------------------
// K1: ori[img,p] = sigmoid(max_c conf[img,c,p])   (vectorized float4)
// ---------------------------------------------------------------------------
__global__ __launch_bounds__(256) void k_sigmax(const float* __restrict__ conf,
                                                float* __restrict__ ori) {
  int i = blockIdx.x * 256 + threadIdx.x;   // over BNHW/4 (exact)
  int e = i * 4;
  int img = e / HWP;
  int p   = e % HWP;
  const float4 a = *(const float4*)(conf + ((size_t)img * CCH    ) * HWP + p);
  const float4 b = *(const float4*)(conf + ((size_t)img * CCH + 1) * HWP + p);
  float4 o;
  o.x = 1.f / (1.f + expf(-fmaxf(a.x, b.x)));
  o.y = 1.f / (1.f + expf(-fmaxf(a.y, b.y)));
  o.z = 1.f / (1.f + expf(-fmaxf(a.z, b.z)));
  o.w = 1.f / (1.f + expf(-fmaxf(a.w, b.w)));
  *(float4*)(ori + (size_t)img * HWP + p) = o;
}

// banded-matrix weight select: returns w[d] for d in [0,4], else 0
__device__ __forceinline__ float band(int d, float w0, float w1, float w2,
                                      float w3, float w4) {
  float r = 0.f;
  r = (d == 0) ? w0 : r;
  r = (d == 1) ? w1 : r;
  r = (d == 2) ? w2 : r;
  r = (d == 3) ? w3 : r;
  r = (d == 4) ? w4 : r;
  return r;
}

// ---------------------------------------------------------------------------
// K2: 5x5 SAME conv on the matrix pipe. Each wave computes a 16x16 output
//     tile as 25 accumulating V_WMMA_F32_16X16X4_F32 ops:
//       out = sum_{dy,kk} A_{dy,kk}[16x4] x B_{dy,kk}[4x16]
//     A = shifted column-slices of the LDS halo (f32, exact), B = banded
//     matrix of kernel taps built from SGPR weights. Halo (132x20) is staged
//     with CDNA5 async global->LDS copies (ASYNCcnt); OOB zero-filled via DS.
// ---------------------------------------------------------------------------
__global__ __launch_bounds__(256) void k_conv5_wmma(const float* __restrict__ ori,
                                                    const float* __restrict__ kern,
                                                    float* __restrict__ comm) {
  __shared__ float lds[20 * LSTR];
  const int img = blockIdx.z;
  const int bx = blockIdx.x * 128;
  const int by = blockIdx.y * 16;
  const float* __restrict__ src = ori + (size_t)img * HWP;
  const int tid = threadIdx.x;

  // 25 taps -> SGPRs (uniform address, constant offsets)
  float w[25];
  #pragma unroll
  for (int i = 0; i < 25; ++i) w[i] = kern[i];

  // stage 132x20 input halo
  for (int e = tid; e < 20 * 132; e += 256) {
    int ly = e / 132, lx = e % 132;
    int gy = by + ly - 2, gx = bx + lx - 2;
    unsigned lds_off = (unsigned)(unsigned long long)(&lds[ly * LSTR + lx]);
    if ((gx >= 0) & (gx < WW) & (gy >= 0) & (gy < HH)) {
      const float* g = src + gy * WW + gx;
      asm volatile("global_load_async_to_lds_b32 %0, %1, off"
                   :: "v"(lds_off), "v"(g) : "memory");
    } else {
      lds[ly * LSTR + lx] = 0.f;
    }
  }
  asm volatile("s_wait_asynccnt 0x0" ::: "memory");
  __syncthreads();

  const int lane  = tid & 31;
  const int wv    = tid >> 5;        // wave id = 16-wide x-subtile in block
  const int half  = lane >> 4;       // 0 -> K={0,1}, 1 -> K={2,3}  (ISA A/B layout)
  const int mn    = lane & 15;       // A row m  /  B,D column n
  const int kbase = half * 2;

  v8f c = {};
  #pragma unroll
  for (int dy = 0; dy < 5; ++dy) {
    #pragma unroll
    for (int kk = 0; kk < 5; ++kk) {
      // A[m,k] = in[m+dy, wv*16 + kk*4 + k]  (LDS col 0 == global x0-2)
      const int col = wv * 16 + kk * 4 + kbase;
      const float2 av = *(const float2*)&lds[(mn + dy) * LSTR + col];
      v2f a; a.x = av.x; a.y = av.y;
      // B[k,n] = w[dy, kk*4+k-n] banded
      const int d0 = kk * 4 + kbase - mn;
      v2f b;
      b.x = band(d0,     w[dy*5+0], w[dy*5+1], w[dy*5+2], w[dy*5+3], w[dy*5+4]);
      b.y = band(d0 + 1, w[dy*5+0], w[dy*5+1], w[dy*5+2], w[dy*5+3], w[dy*5+4]);
      c = __builtin_amdgcn_wmma_f32_16x16x4_f32(false, a, false, b,
                                                (short)0, c, false, false);
    }
  }

  // D 16x16 f32 layout: VGPR j = row j (lanes 0-15) / row j+8 (lanes 16-31)
  const int tx0 = bx + wv * 16;
  float* __restrict__ dst = comm + (size_t)img * HWP;
  #pragma unroll
  for (int j = 0; j < 8; ++j) {
    const int row = half ? (j + 8) : j;
    dst[(by + row) * WW + tx0 + mn] = c[j];
  }
}

// ---------------------------------------------------------------------------
// Bilinear sample with zero padding (matches reference _warp_affine corners)
// ---------------------------------------------------------------------------
__device__ __forceinline__ float fetchz(const float* __restrict__ im, float yy, float xx) {
  if (xx < 0.f || xx > (float)(WW - 1) || yy < 0.f || yy > (float)(HH - 1)) return 0.f;
  return im[(int)yy * WW + (int)xx];
}
__device__ __forceinline__ float bilin(const float* __restrict__ im, float cx, float cy) {
  float ix = ((cx + 1.f) * (float)WW - 1.f) * 0.5f;
  float iy = ((cy + 1.f) * (float)HH - 1.f) * 0.5f;
  float x0 = floorf(ix), y0 = floorf(iy);
  float wx1 = ix - x0, wx0 = 1.f - wx1;
  float wy1 = iy - y0, wy0 = 1.f - wy1;
  float acc = fetchz(im, y0,       x0      ) * (wy0 * wx0);
  acc      += fetchz(im, y0,       x0 + 1.f) * (wy0 * wx1);
  acc      += fetchz(im, y0 + 1.f, x0      ) * (wy1 * wx0);
  acc      += fetchz(im, y0 + 1.f, x0 + 1.f) * (wy1 * wx1);
  return acc;
}

// ---------------------------------------------------------------------------
// K3: per pixel p of batch b: warp all 8 comm maps with theta[b,0,n], then
//     stable-descending sort of agents 1..7 (adjacent-swap network => stable),
//     cumsum starting at agent 0, mask = cum < 2.0, unsort -> ego_mask.
// ---------------------------------------------------------------------------
__global__ __launch_bounds__(256) void k_egomask(const float* __restrict__ comm,
                                                 const float* __restrict__ theta,
                                                 float* __restrict__ egoMask) {
  const int p = blockIdx.x * 256 + threadIdx.x;  // HWP = 576*256 exact
  const int b = blockIdx.y;
  const int px = p % WW, py = p / WW;
  const float xs = (2.f * px + 1.f) / (float)WW - 1.f;
  const float ys = (2.f * py + 1.f) / (float)HH - 1.f;

  float v[NAG];
  #pragma unroll
  for (int n = 0; n < NAG; ++n) {
    const float* t = theta + ((size_t)b * NAG * NAG + n) * 6;   // t_mat[b][0][n]
    float cx = t[0] * xs + t[1] * ys + t[2];
    float cy = t[3] * xs + t[4] * ys + t[5];
    v[n] = bilin(comm + ((size_t)b * NAG + n) * HWP, cx, cy);
  }

  float r[7]; int id[7];
  #pragma unroll
  for (int k = 0; k < 7; ++k) { r[k] = v[k + 1]; id[k] = k; }
  // bubble network, adjacent swaps with strict '<' => stable descending
  #pragma unroll
  for (int i = 0; i < 6; ++i)
    #pragma unroll
    for (int j = 0; j < 6 - i; ++j)
      if (r[j] < r[j + 1]) {
        float tv = r[j]; r[j] = r[j + 1]; r[j + 1] = tv;
        int   ti = id[j]; id[j] = id[j + 1]; id[j + 1] = ti;
      }

  float cum = v[0];
  egoMask[((size_t)b * NAG) * HWP + p] = (cum < 2.0f) ? 1.f : 0.f;
  #pragma unroll
  for (int k = 0; k < 7; ++k) {
    cum += r[k];
    float m = (cum < 2.0f) ? 1.f : 0.f;
    egoMask[((size_t)b * NAG + 1 + id[k]) * HWP + p] = m;
  }
}

// ---------------------------------------------------------------------------
// K4: communication_mask = bilin(ego_mask[b,n], theta[b,n,0]) * (comm>0.5);
//     per-block partial sums of agents>=1 (deterministic, no atomics);
//     agent 0 plane forced to 1; writes masks and ori*mask in place.
// ---------------------------------------------------------------------------
__global__ __launch_bounds__(256) void k_final(const float* __restrict__ egoMask,
                                               const float* __restrict__ theta,
                                               float* __restrict__ out,       // d_out base
                                               float* __restrict__ partials) {
  const int p   = blockIdx.x * 256 + threadIdx.x;
  const int img = blockIdx.y;                 // b*NAG + n
  const int b = img >> 3, n = img & 7;
  const int px = p % WW, py = p / WW;
  const float xs = (2.f * px + 1.f) / (float)WW - 1.f;
  const float ys = (2.f * py + 1.f) / (float)HH - 1.f;

  const float* t = theta + (((size_t)b * NAG + n) * NAG) * 6;   // t_mat[b][n][0]
  float cx = t[0] * xs + t[1] * ys + t[2];
  float cy = t[3] * xs + t[4] * ys + t[5];
  float wm = bilin(egoMask + (size_t)img * HWP, cx, cy);

  const size_t idx = (size_t)img * HWP + p;
  float commv = out[BNHW + idx];                    // comm plane (read own index)
  float cm = ((commv > 0.5f) ? 1.f : 0.f) * wm;
  float contrib = (n > 0) ? cm : 0.f;

  // wave32 shuffle reduce + cross-wave LDS reduce (fixed order -> deterministic)
  float s = contrib;
  #pragma unroll
  for (int o = 16; o > 0; o >>= 1) s += __shfl_down(s, o, 32);
  __shared__ float red[8];
  if ((threadIdx.x & 31) == 0) red[threadIdx.x >> 5] = s;
  __syncthreads();
  if (threadIdx.x == 0) {
    float tsum = 0.f;
    #pragma unroll
    for (int wv = 0; wv < 8; ++wv) tsum += red[wv];
    partials[(size_t)blockIdx.y * gridDim.x + blockIdx.x] = tsum;
  }

  float mout = (n == 0) ? 1.f : cm;
  out[BNHW + idx] = mout;                           // communication_masks
  out[idx] = out[idx] * mout;                       // batch_maps = ori * mask
}

// ---------------------------------------------------------------------------
// K5: deterministic tree reduction of block partials -> scalar rate
// ---------------------------------------------------------------------------
__global__ __launch_bounds__(256) void k_rate(const float* __restrict__ partials,
                                              int nPart, float* __restrict__ rate_out) {
  __shared__ float red[256];
  float s = 0.f;
  for (int i = threadIdx.x; i < nPart; i += 256) s += partials[i];
  red[threadIdx.x] = s;
  __syncthreads();
  for (int o = 128; o > 0; o >>= 1) {
    if (threadIdx.x < (unsigned)o) red[threadIdx.x] += red[threadIdx.x + o];
    __syncthreads();
  }
  if (threadIdx.x == 0)
    rate_out[0] = red[0] / (float)(HWP * (NAG - 1)) / (float)BB;
}

// ---------------------------------------------------------------------------
extern "C" void kernel_launch(void* const* d_in, const int* in_sizes, int n_in,
                              void* d_out, int out_size, void* d_ws, size_t ws_size,
                              hipStream_t stream) {
  (void)in_sizes; (void)n_in; (void)out_size; (void)ws_size;
  const float* conf  = (const float*)d_in[0];
  // d_in[1] = record_len (unused; always N)
  const float* theta = (const float*)d_in[2];
  const float* kern  = (const float*)d_in[3];

  float* out  = (float*)d_out;
  float* ori  = out;                 // reuse batch_maps plane as ori scratch
  float* comm = out + (size_t)BNHW;  // reuse masks plane as comm scratch
  float* ego      = (float*)d_ws;                    // BNHW floats (~37.7 MB)
  float* partials = ego + (size_t)BNHW;              // 576*64 floats

  k_sigmax <<< BNHW / 4 / 256, 256, 0, stream >>> (conf, ori);

  dim3 g2(WW / 128, HH / 16, BB * NAG);              // (3, 24, 64)
  k_conv5_wmma <<< g2, 256, 0, stream >>> (ori, kern, comm);

  k_egomask<<< dim3(HWP / 256, BB), 256, 0, stream >>> (comm, theta, ego);

  k_final  <<< dim3(HWP / 256, BB * NAG), 256, 0, stream >>> (ego, theta, out, partials);

  k_rate   <<< 1, 256, 0, stream >>> (partials, (HWP / 256) * BB * NAG,
                                      out + 2 * (size_t)BNHW);
}